// TCRgnnEdgeLayers_22720376996121
// MI455X (gfx1250) — compile-verified
//
#include <hip/hip_runtime.h>

#define H 128

typedef __attribute__((ext_vector_type(2))) float v2f;
typedef __attribute__((ext_vector_type(8))) float v8f;

// ---------------------------------------------------------------------------
// Edge phase: m = relu(x[src] + edge_attr[e]*We + be); agg[dst] += m
// One wave (32 lanes) per edge, 4 consecutive floats per lane.
// x and agg are 51MB each -> gathers and f32 atomics stay in the 192MB L2.
// ---------------------------------------------------------------------------
__global__ void edge_scatter_kernel(const float* __restrict__ x,
                                    const int* __restrict__ ei,
                                    const float* __restrict__ ea,
                                    const float* __restrict__ We,
                                    const float* __restrict__ be,
                                    float* __restrict__ agg,
                                    int E) {
  long t = (long)blockIdx.x * blockDim.x + threadIdx.x;
  int e = (int)(t >> 5);
  int lane = (int)(t & 31);
  if (e >= E) return;
  int src = ei[e];
  int dst = ei[E + e];
  float w = ea[e];
  int h0 = lane * 4;
  const float4 xs = *(const float4*)(x + (size_t)src * H + h0);
  const float4 wv = *(const float4*)(We + h0);
  const float4 bv = *(const float4*)(be + h0);
  float m0 = fmaxf(xs.x + fmaf(w, wv.x, bv.x), 0.0f);
  float m1 = fmaxf(xs.y + fmaf(w, wv.y, bv.y), 0.0f);
  float m2 = fmaxf(xs.z + fmaf(w, wv.z, bv.z), 0.0f);
  float m3 = fmaxf(xs.w + fmaf(w, wv.w, bv.w), 0.0f);
  float* dp = agg + (size_t)dst * H + h0;
  atomicAdd(dp + 0, m0);
  atomicAdd(dp + 1, m1);
  atomicAdd(dp + 2, m2);
  atomicAdd(dp + 3, m3);
}

// ---------------------------------------------------------------------------
// FP32 WMMA GEMM: C[M,H] = relu( (A (+AG)) [M,H] @ B[H,H] + bias[H] )
// One wave computes a 32x32 output block (2x2 grid of 16x16 WMMA tiles):
// per K-step of 4, 2 A-loads + 2 B-loads feed 4x V_WMMA_F32_16X16X4_F32.
// ADD is a compile-time flag -> no divergent pointer test in the hot loop.
// A 16x4 layout (ISA 7.12.2): lanes 0-15 -> row M=l16, v0=K+0, v1=K+1;
//                             lanes 16-31 -> row M=l16, v0=K+2, v1=K+3.
// B 4x16 mirrors A with M<->N. C/D: VGPR r -> M=r / r+8, N=l16.
// ---------------------------------------------------------------------------
template <bool ADD>
__global__ void gemm_wmma_kernel(const float* __restrict__ A,
                                 const float* __restrict__ AG,
                                 const float* __restrict__ B,
                                 const float* __restrict__ bias,
                                 float* __restrict__ C,
                                 int tilesM2) {  // = (M/16)/2
  int wave = threadIdx.x >> 5;
  int lane = threadIdx.x & 31;
  int tile = blockIdx.x * (blockDim.x >> 5) + wave;
  int tiles = tilesM2 * 4;       // (H/16)/2 == 4 column pairs
  if (tile >= tiles) return;     // wave-uniform: EXEC stays all-ones
  int tm2 = tile >> 2;
  int tn2 = tile & 3;
  bool hi = lane >= 16;
  int l16 = lane & 15;
  int arow0 = tm2 * 32 + l16;
  int bcol0 = tn2 * 32 + l16;
  int kofs = hi ? 2 : 0;

  const float* ap0 = A + (size_t)arow0 * H + kofs;
  const float* ap1 = ap0 + 16 * H;
  const float* agp0 = ADD ? (AG + (size_t)arow0 * H + kofs) : nullptr;
  const float* agp1 = ADD ? (agp0 + 16 * H) : nullptr;
  const float* bp = B + bcol0;

  v8f acc00 = {}, acc01 = {}, acc10 = {}, acc11 = {};
  for (int k = 0; k < H; k += 4) {
    v2f a0, a1, b0, b1;
    a0.x = ap0[k];
    a0.y = ap0[k + 1];
    a1.x = ap1[k];
    a1.y = ap1[k + 1];
    if (ADD) {
      a0.x += agp0[k];
      a0.y += agp0[k + 1];
      a1.x += agp1[k];
      a1.y += agp1[k + 1];
    }
    int ka = k + kofs;
    b0.x = bp[(size_t)ka * H];
    b0.y = bp[(size_t)(ka + 1) * H];
    b1.x = bp[(size_t)ka * H + 16];
    b1.y = bp[(size_t)(ka + 1) * H + 16];
    acc00 = __builtin_amdgcn_wmma_f32_16x16x4_f32(false, a0, false, b0,
                                                  (short)0, acc00, false, false);
    acc01 = __builtin_amdgcn_wmma_f32_16x16x4_f32(false, a0, false, b1,
                                                  (short)0, acc01, false, false);
    acc10 = __builtin_amdgcn_wmma_f32_16x16x4_f32(false, a1, false, b0,
                                                  (short)0, acc10, false, false);
    acc11 = __builtin_amdgcn_wmma_f32_16x16x4_f32(false, a1, false, b1,
                                                  (short)0, acc11, false, false);
  }

  float bi0 = bias[bcol0];
  float bi1 = bias[bcol0 + 16];
  int mbase0 = tm2 * 32 + (hi ? 8 : 0);
  int mbase1 = mbase0 + 16;
  #pragma unroll
  for (int r = 0; r < 8; ++r) {
    float* c0 = C + (size_t)(mbase0 + r) * H + bcol0;
    float* c1 = C + (size_t)(mbase1 + r) * H + bcol0;
    c0[0]  = fmaxf(acc00[r] + bi0, 0.0f);
    c0[16] = fmaxf(acc01[r] + bi1, 0.0f);
    c1[0]  = fmaxf(acc10[r] + bi0, 0.0f);
    c1[16] = fmaxf(acc11[r] + bi1, 0.0f);
  }
}

// ---------------------------------------------------------------------------
// Mean-pool scatter: pooled[batch[n]] += x[n]; cnt[batch[n]] += 1
// ---------------------------------------------------------------------------
__global__ void pool_scatter_kernel(const float* __restrict__ x,
                                    const int* __restrict__ batch,
                                    float* __restrict__ pooled,
                                    float* __restrict__ cnt,
                                    int N) {
  long t = (long)blockIdx.x * blockDim.x + threadIdx.x;
  int n = (int)(t >> 5);
  int lane = (int)(t & 31);
  if (n >= N) return;
  int g = batch[n];
  int h0 = lane * 4;
  const float4 xs = *(const float4*)(x + (size_t)n * H + h0);
  float* pp = pooled + (size_t)g * H + h0;
  atomicAdd(pp + 0, xs.x);
  atomicAdd(pp + 1, xs.y);
  atomicAdd(pp + 2, xs.z);
  atomicAdd(pp + 3, xs.w);
  if (lane == 0) atomicAdd(cnt + g, 1.0f);
}

__global__ void pool_div_kernel(float* __restrict__ pooled,
                                const float* __restrict__ cnt, int GH) {
  int t = blockIdx.x * blockDim.x + threadIdx.x;
  if (t >= GH) return;
  int g = t / H;
  pooled[t] = pooled[t] / fmaxf(cnt[g], 1.0f);
}

// ---------------------------------------------------------------------------
// Final projection: out[g] = dot(clsh[g,:], Wc2) + bc2. One wave per graph.
// ---------------------------------------------------------------------------
__global__ void final_dot_kernel(const float* __restrict__ clsh,
                                 const float* __restrict__ Wc2,
                                 const float* __restrict__ bc2,
                                 float* __restrict__ out, int G) {
  long t = (long)blockIdx.x * blockDim.x + threadIdx.x;
  int g = (int)(t >> 5);
  int lane = (int)(t & 31);
  if (g >= G) return;
  int h0 = lane * 4;
  const float4 p = *(const float4*)(clsh + (size_t)g * H + h0);
  const float4 w = *(const float4*)(Wc2 + h0);
  float s = p.x * w.x + p.y * w.y + p.z * w.z + p.w * w.w;
  #pragma unroll
  for (int off = 16; off > 0; off >>= 1) s += __shfl_xor(s, off, 32);
  if (lane == 0) out[g] = s + bc2[0];
}

// ---------------------------------------------------------------------------
extern "C" void kernel_launch(void* const* d_in, const int* in_sizes, int n_in,
                              void* d_out, int out_size, void* d_ws, size_t ws_size,
                              hipStream_t stream) {
  const float* x    = (const float*)d_in[0];
  const int*   ei   = (const int*)d_in[1];
  const float* ea   = (const float*)d_in[2];
  const int*   bat  = (const int*)d_in[3];
  const float* We   = (const float*)d_in[4];
  const float* be   = (const float*)d_in[5];
  const float* W1   = (const float*)d_in[6];
  const float* b1   = (const float*)d_in[7];
  const float* W2   = (const float*)d_in[8];
  const float* b2   = (const float*)d_in[9];
  const float* Wc1  = (const float*)d_in[10];
  const float* bc1  = (const float*)d_in[11];
  const float* Wc2  = (const float*)d_in[12];
  const float* bc2  = (const float*)d_in[13];
  float* out = (float*)d_out;

  const int N = in_sizes[0] / H;
  const int E = in_sizes[2];
  const int G = out_size;          // output is [G,1]
  const int L = in_sizes[5] / H;   // be is [L,H]

  char* ws = (char*)d_ws;
  float* xcur   = (float*)ws; ws += (size_t)N * H * sizeof(float);
  float* agg    = (float*)ws; ws += (size_t)N * H * sizeof(float);
  float* tbuf   = (float*)ws; ws += (size_t)N * H * sizeof(float);
  float* pooled = (float*)ws; ws += (size_t)G * H * sizeof(float);
  float* cnt    = (float*)ws; ws += (size_t)G * sizeof(float);
  float* clsh   = (float*)ws;

  // working copy of node features (inputs must not be mutated)
  hipMemcpyAsync(xcur, x, (size_t)N * H * sizeof(float),
                 hipMemcpyDeviceToDevice, stream);

  const int tilesM2 = (N / 16) / 2;               // 3125 (N % 32 == 0)
  const int tiles   = tilesM2 * 4;
  const int gblocks = (tiles + 7) / 8;            // 8 waves per 256-thr block
  const long ethr   = (long)E * 32;
  const int eblocks = (int)((ethr + 255) / 256);

  for (int l = 0; l < L; ++l) {
    hipMemsetAsync(agg, 0, (size_t)N * H * sizeof(float), stream);
    edge_scatter_kernel<<<eblocks, 256, 0, stream>>>(
        xcur, ei, ea, We + (size_t)l * H, be + (size_t)l * H, agg, E);
    // t = relu((x + agg) @ W1 + b1)
    gemm_wmma_kernel<true><<<gblocks, 256, 0, stream>>>(
        xcur, agg, W1 + (size_t)l * H * H, b1 + (size_t)l * H, tbuf, tilesM2);
    // x = relu(t @ W2 + b2)   (MLP output + post-conv ReLU fused)
    gemm_wmma_kernel<false><<<gblocks, 256, 0, stream>>>(
        tbuf, nullptr, W2 + (size_t)l * H * H, b2 + (size_t)l * H, xcur, tilesM2);
  }

  // mean pool per graph
  hipMemsetAsync(pooled, 0, (size_t)G * H * sizeof(float), stream);
  hipMemsetAsync(cnt, 0, (size_t)G * sizeof(float), stream);
  {
    long pthr = (long)N * 32;
    pool_scatter_kernel<<<(int)((pthr + 255) / 256), 256, 0, stream>>>(
        xcur, bat, pooled, cnt, N);
    int gh = G * H;
    pool_div_kernel<<<(gh + 255) / 256, 256, 0, stream>>>(pooled, cnt, gh);
  }

  // classifier: clsh = relu(pooled @ Wc1 + bc1);  out = clsh @ Wc2 + bc2
  {
    int ctilesM2 = (G / 16) / 2;                  // 16 (G % 32 == 0)
    int ctiles = ctilesM2 * 4;
    gemm_wmma_kernel<false><<<(ctiles + 7) / 8, 256, 0, stream>>>(
        pooled, nullptr, Wc1, bc1, clsh, ctilesM2);
    long fthr = (long)G * 32;
    final_dot_kernel<<<(int)((fthr + 255) / 256), 256, 0, stream>>>(
        clsh, Wc2, bc2, out, G);
  }
}